// OrdinalModifiedRegression_63230508532389
// MI455X (gfx1250) — compile-verified
//
#include <hip/hip_runtime.h>
#include <stdint.h>

// OrdinalModifiedRegression reduces exactly to:
//   d = |argmax(pred_row) - target|; if (d == 1) d = 0; out = mean(d)
// Streaming reduction: 128MB @ 23.3TB/s ~= 5.6us floor. CDNA5 async data
// mover (global_load_async_to_lds_b128, ASYNCcnt) as a 4-deep pipeline:
// one wave per row, one float4 per lane, ~32MB in flight machine-wide.

#define WAVES_PER_BLOCK 8
#define THREADS (WAVES_PER_BLOCK * 32)
#define NBLOCKS 2048
#define TOTAL_WAVES (NBLOCKS * WAVES_PER_BLOCK)   // 16384, power of two
#define PIPE 4                                    // async pipeline depth

__global__ void __launch_bounds__(THREADS)
ord_argmax_dist_kernel(const float* __restrict__ pred,
                       const long long* __restrict__ tgt,
                       unsigned* __restrict__ partials,
                       int B)
{
    // 8 waves * 4 buffers * 128 floats = 16KB LDS
    __shared__ float buf[WAVES_PER_BLOCK * PIPE * 128];
    __shared__ unsigned wpart[WAVES_PER_BLOCK];

    const int lane = threadIdx.x & 31;
    const int wid  = threadIdx.x >> 5;
    const int gw   = (int)blockIdx.x * WAVES_PER_BLOCK + wid;

    // Low 32 bits of a flat pointer into LDS == LDS byte address.
    const uint32_t ldsWaveBase =
        (uint32_t)(uintptr_t)(&buf[0]) + (uint32_t)wid * (PIPE * 512u);
    const uint64_t gbase = (uint64_t)(uintptr_t)pred;

    // TOTAL_WAVES is a compile-time power of two -> shifts, no division.
    const int n = (gw < B) ? ((B - 1 - gw) / TOTAL_WAVES + 1) : 0;

    unsigned acc = 0;

    // Issue: each of 32 lanes async-copies 16B -> one 512B row into LDS.
    auto issue = [&](int k) {
        uint32_t row   = (uint32_t)(gw + k * TOTAL_WAVES);
        uint32_t goff  = row * 512u + (uint32_t)lane * 16u;        // C = 128 floats
        uint32_t laddr = ldsWaveBase + (uint32_t)(k & (PIPE - 1)) * 512u
                       + (uint32_t)lane * 16u;
        asm volatile("global_load_async_to_lds_b128 %0, %1, %2"
                     :
                     : "v"(laddr), "v"(goff), "s"(gbase)
                     : "memory");
    };

    // Prologue: fill the pipeline (up to PIPE-1 ahead).
    #pragma unroll
    for (int k = 0; k < PIPE - 1; ++k)
        if (k < n) issue(k);

    for (int k = 0; k < n; ++k) {
        if (k + PIPE - 1 < n) {
            issue(k + PIPE - 1);
            // loads complete in order: <=PIPE-1 outstanding => row k resident
            asm volatile("s_wait_asynccnt 3" ::: "memory");
        } else {
            const int rem = n - 1 - k;   // rows still outstanding beyond k
            if (rem >= 2)      asm volatile("s_wait_asynccnt 2" ::: "memory");
            else if (rem == 1) asm volatile("s_wait_asynccnt 1" ::: "memory");
            else               asm volatile("s_wait_asynccnt 0" ::: "memory");
        }

        const float4 v = *(const float4*)(
            &buf[wid * (PIPE * 128) + (k & (PIPE - 1)) * 128 + lane * 4]);

        // per-lane argmax over 4 columns (strict > keeps first occurrence)
        float m  = v.x;
        int   id = lane * 4;
        if (v.y > m) { m = v.y; id = lane * 4 + 1; }
        if (v.z > m) { m = v.z; id = lane * 4 + 2; }
        if (v.w > m) { m = v.w; id = lane * 4 + 3; }

        // wave32 butterfly, min-index tie-break (matches jnp.argmax)
        #pragma unroll
        for (int s = 16; s > 0; s >>= 1) {
            float om = __shfl_xor(m, s, 32);
            int   oi = __shfl_xor(id, s, 32);
            if (om > m || (om == m && oi < id)) { m = om; id = oi; }
        }

        if (lane == 0) {
            int t = (int)tgt[gw + k * TOTAL_WAVES];
            int d = id - t;
            d = d < 0 ? -d : d;
            if (d == 1) d = 0;      // +-1 snap-to-label => zero distance
            acc += (unsigned)d;
        }
    }

    if (lane == 0) wpart[wid] = acc;
    __syncthreads();
    if (threadIdx.x == 0) {
        unsigned s = 0;
        #pragma unroll
        for (int i = 0; i < WAVES_PER_BLOCK; ++i) s += wpart[i];
        partials[blockIdx.x] = s;   // every slot written every call: no init needed
    }
}

__global__ void __launch_bounds__(256)
ord_finish_kernel(const unsigned* __restrict__ partials, int nPart,
                  float* __restrict__ out, int B)
{
    __shared__ unsigned long long smem[256];
    unsigned long long a = 0;
    for (int i = threadIdx.x; i < nPart; i += 256) a += (unsigned long long)partials[i];
    smem[threadIdx.x] = a;
    __syncthreads();
    #pragma unroll
    for (int off = 128; off > 0; off >>= 1) {
        if ((int)threadIdx.x < off) smem[threadIdx.x] += smem[threadIdx.x + off];
        __syncthreads();
    }
    if (threadIdx.x == 0) out[0] = (float)((double)smem[0] / (double)B);
}

extern "C" void kernel_launch(void* const* d_in, const int* in_sizes, int n_in,
                              void* d_out, int out_size, void* d_ws, size_t ws_size,
                              hipStream_t stream)
{
    const float*     pred = (const float*)d_in[0];
    const long long* tgt  = (const long long*)d_in[1];  // int64 targets
    const int B = in_sizes[1];                          // 262144 (C fixed at 128)

    unsigned* partials = (unsigned*)d_ws;               // NBLOCKS * 4 bytes

    ord_argmax_dist_kernel<<<NBLOCKS, THREADS, 0, stream>>>(pred, tgt, partials, B);
    ord_finish_kernel<<<1, 256, 0, stream>>>(partials, NBLOCKS, (float*)d_out, B);
}